// UnicycleIntegrator_88837103551333
// MI455X (gfx1250) — compile-verified
//
#include <hip/hip_runtime.h>
#include <hip/hip_bf16.h>

// ---------------------------------------------------------------------------
// MI455X (gfx1250) implementation.
//
// Problem: B=4096 trajectories, T=64.
//   theta: constant-SPD solve -> precompute 64x64 inverse once, then batched
//          inverse-apply as an f32 GEMM on the matrix unit
//          (V_WMMA_F32_16X16X4_F32; f32 required: lhs spans ~1e7 in magnitude).
//   v:     per-batch band-3 (7-diagonal) SPD 63x63 system -> one batch per
//          lane, band Cholesky with the factor held in LDS.
//
// Memory-bound: ~15 MB total traffic -> ~0.65 us at 23.3 TB/s. Compute is
// ~40 MFLOP, far below the WMMA ceiling, so the structure above (single
// streaming pass + tiny setup kernel) is the optimum.
// ---------------------------------------------------------------------------

#define BATCH 4096
#define TT    64          // T
#define NVV   63          // v-system size (T-1)
#define C6    1.0e6f      // lambda / dt^6  (lambda = 1, dt = 0.1)
#define RIDGE 1e-6f
#define TWO_OVER_DT 20.0f // 2/dt

typedef float v2f __attribute__((ext_vector_type(2)));
typedef float v8f __attribute__((ext_vector_type(8)));

// (D3^T D3)[i][j] for an (rows+3)-point 3rd-order FD operator with `rows` rows.
__device__ __forceinline__ float dtd3s(int i, int j, int rows) {
    const float cf[4] = {-1.f, 3.f, -3.f, 1.f};
    int lo = i - 3; if (j - 3 > lo) lo = j - 3; if (lo < 0) lo = 0;
    int hi = i;     if (j < hi)     hi = j;     if (hi > rows - 1) hi = rows - 1;
    float s = 0.f;
    for (int r = lo; r <= hi; ++r) s += cf[i - r] * cf[j - r];
    return s;
}

// ---------------------------------------------------------------------------
// Kernel 1: build lhs_theta = (1+ridge) I + C6 * DTD65[1:,1:]  (band-3 SPD),
// band-Cholesky it (serial, trivial), then 64 threads each back out one
// column of the inverse via band forward/back substitution. ~0.5 MFLOP total.
// ---------------------------------------------------------------------------
__global__ void setup_minv_kernel(float* __restrict__ Minv) {
    __shared__ float Lb[64][4];     // Lb[j][d] = L[j][j-d]
    __shared__ float zz[64 * 64];   // zz[i*64 + t]: column t workspace
    const int t = threadIdx.x;

    if (t == 0) {
        for (int j = 0; j < 64; ++j) {
            int lo = j - 3; if (lo < 0) lo = 0;
            for (int k = lo; k < j; ++k) {
                float a = C6 * dtd3s(j + 1, k + 1, 62);
                for (int m = lo; m < k; ++m)
                    a -= Lb[j][j - m] * Lb[k][k - m];
                Lb[j][j - k] = a / Lb[k][0];
            }
            float a = 1.0f + RIDGE + C6 * dtd3s(j + 1, j + 1, 62);
            for (int m = lo; m < j; ++m) { float l = Lb[j][j - m]; a -= l * l; }
            Lb[j][0] = sqrtf(a);
        }
    }
    __syncthreads();

    // forward: L z = e_t
    for (int i = 0; i < 64; ++i) {
        float s = (i == t) ? 1.0f : 0.0f;
        int dmax = i < 3 ? i : 3;
        for (int d = 1; d <= dmax; ++d) s -= Lb[i][d] * zz[(i - d) * 64 + t];
        zz[i * 64 + t] = s / Lb[i][0];
    }
    // backward: L^T x = z  (in place)
    for (int i = 63; i >= 0; --i) {
        float s = zz[i * 64 + t];
        int dmax = 63 - i; if (dmax > 3) dmax = 3;
        for (int d = 1; d <= dmax; ++d) s -= Lb[i + d][d] * zz[(i + d) * 64 + t];
        zz[i * 64 + t] = s / Lb[i][0];
    }
    for (int i = 0; i < 64; ++i) Minv[i * 64 + t] = zz[i * 64 + t]; // row-major inverse
}

// ---------------------------------------------------------------------------
// Kernel 2: yaw extraction + angle unwrap. One block per batch, 64 threads.
// Writes unwrapped theta into the v-slot of d_out (staging; overwritten later).
// ---------------------------------------------------------------------------
__global__ void theta_prep_kernel(const float* __restrict__ rot,
                                  float* __restrict__ out) {
    const int b = blockIdx.x, t = threadIdx.x;
    const float* rb = rot + (size_t)b * (TT * 9);
    float phi = atan2f(rb[t * 9 + 3], rb[t * 9 + 0]);   // atan2(R10, R00)

    __shared__ float sh[64];
    __shared__ float sc[64];
    sh[t] = phi;
    __syncthreads();
    float d;
    if (t == 0) d = phi;
    else { float dd = phi - sh[t - 1]; d = atan2f(sinf(dd), cosf(dd)); }
    sc[t] = d;
    __syncthreads();
    // Hillis-Steele inclusive scan -> unwrapped theta
    for (int off = 1; off < 64; off <<= 1) {
        float v = sc[t];
        if (t >= off) v += sc[t - off];
        __syncthreads();
        sc[t] = v;
        __syncthreads();
    }
    out[(size_t)b * 128 + 64 + t] = sc[t];
}

// ---------------------------------------------------------------------------
// Kernel 3: theta_smooth = Minv @ theta_u, as [4096x64]x[64x64] f32 GEMM.
// One wave per 16 batches; 16 K-chunks x 4 N-tiles = 64 V_WMMA_F32_16X16X4_F32.
// A[m][k] = theta_u[b0+m][k];  B[k][n] = Minv[t0+n][k] (Minv symmetric).
// ISA fragment layouts: A/B element e of lane l holds k = 2*(l>>4)+e;
// D vgpr r of lane l holds row r + 8*(l>>4), col l&15.
// ---------------------------------------------------------------------------
__global__ void theta_gemm_kernel(const float* __restrict__ Minv,
                                  float* __restrict__ out) {
    const int lane = threadIdx.x;
    const int l16  = lane & 15;
    const int half = lane >> 4;
    const int b0   = blockIdx.x * 16;

    v8f acc[4] = {};
    const float* aRow = out + (size_t)(b0 + l16) * 128 + 64;  // theta_u of this batch row

    #pragma unroll
    for (int k0 = 0; k0 < 64; k0 += 4) {
        const int ka = k0 + 2 * half;
        v2f a = *reinterpret_cast<const v2f*>(aRow + ka);
        #pragma unroll
        for (int nt = 0; nt < 4; ++nt) {
            const int t0 = nt * 16;
            v2f bf = *reinterpret_cast<const v2f*>(Minv + (size_t)(t0 + l16) * 64 + ka);
            acc[nt] = __builtin_amdgcn_wmma_f32_16x16x4_f32(
                false, a, false, bf, (short)0, acc[nt], false, false);
        }
    }
    #pragma unroll
    for (int nt = 0; nt < 4; ++nt) {
        const int t0 = nt * 16;
        #pragma unroll
        for (int r = 0; r < 8; ++r) {
            const int m = r + 8 * half;
            out[(size_t)(b0 + m) * 128 + t0 + l16] = acc[nt][r];
        }
    }
}

// ---------------------------------------------------------------------------
// Kernel 4: per-batch v solve. One lane = one batch; 32 lanes per block.
// lhs = ATA(theta) + C6*DTD64[1:,1:] + ridge I : band-3 SPD 63x63.
// ATA: diag = {2,...,2,1} (rows j=1..63), offdiag = cos(dtheta). Band Cholesky
// with the factor + solution vector in LDS (48.5 KB/block), rhs fused into the
// forward sweep.
// ---------------------------------------------------------------------------
__global__ void v_solve_kernel(const float* __restrict__ xy_all,
                               const float* __restrict__ v0_all,
                               float* __restrict__ out) {
    __shared__ float LbS[NVV * 4 * 32];
    __shared__ float yS [NVV * 32];
    __shared__ float thS[TT * 32];
    const int lane = threadIdx.x;
    const int b = blockIdx.x * 32 + lane;

    const float* th_g = out + (size_t)b * 128;      // smoothed theta (kernel 3)
    const float* xy   = xy_all + (size_t)b * 128;   // [64][2]
    const float v0v   = v0_all[b];

    for (int i = 0; i < TT; ++i) thS[i * 32 + lane] = th_g[i];

    #define LB(p, d) LbS[((p) * 4 + (d)) * 32 + lane]
    #define YV(p)    yS [(p) * 32 + lane]
    #define TH(i)    thS[(i) * 32 + lane]

    for (int p = 0; p < NVV; ++p) {
        int lo = p - 3; if (lo < 0) lo = 0;
        for (int q = lo; q < p; ++q) {
            const int j = p + 1, k = q + 1;
            float a = C6 * dtd3s(j, k, 61);
            if (p == q + 1) a += cosf(TH(j) - TH(k));      // tridiagonal ATA part
            for (int m = lo; m < q; ++m) a -= LB(p, p - m) * LB(q, q - m);
            LB(p, p - q) = a / LB(q, 0);
        }
        const int j = p + 1;
        float a = C6 * dtd3s(j, j, 61) + RIDGE + ((j <= 62) ? 2.0f : 1.0f);
        for (int m = lo; m < p; ++m) { float l = LB(p, p - m); a -= l * l; }
        const float diag = sqrtf(a);
        LB(p, 0) = diag;

        // rhs[j] = cos th_j * (2/dt)(x[j+1]-x[j-1]) + sin th_j * (...) - col0*v0
        int jp = j + 1; if (jp > 63) jp = 63;
        const float sx = TWO_OVER_DT * (xy[2 * jp]     - xy[2 * (j - 1)]);
        const float sy = TWO_OVER_DT * (xy[2 * jp + 1] - xy[2 * (j - 1) + 1]);
        const float th_j = TH(j);
        float r = cosf(th_j) * sx + sinf(th_j) * sy;
        float col0 = C6 * dtd3s(j, 0, 61);
        if (p == 0) col0 += cosf(TH(1) - TH(0));           // ATA[1][0]
        r -= col0 * v0v;

        for (int m = lo; m < p; ++m) r -= LB(p, p - m) * YV(m);  // fused fwd solve
        YV(p) = r / diag;
    }
    for (int p = NVV - 1; p >= 0; --p) {                   // back substitution
        float s = YV(p);
        int dmax = NVV - 1 - p; if (dmax > 3) dmax = 3;
        for (int d = 1; d <= dmax; ++d) s -= LB(p + d, d) * YV(p + d);
        YV(p) = s / LB(p, 0);
    }

    float* vout = out + (size_t)b * 128 + 64;              // overwrite staged theta_u
    vout[0] = v0v;
    for (int p = 0; p < NVV; ++p) vout[1 + p] = YV(p);
    #undef LB
    #undef YV
    #undef TH
}

// ---------------------------------------------------------------------------
extern "C" void kernel_launch(void* const* d_in, const int* in_sizes, int n_in,
                              void* d_out, int out_size, void* d_ws, size_t ws_size,
                              hipStream_t stream) {
    const float* traj_xy  = (const float*)d_in[0];
    const float* traj_rot = (const float*)d_in[1];
    const float* v0       = (const float*)d_in[2];
    float* out  = (float*)d_out;
    float* Minv = (float*)d_ws;     // 64*64 floats = 16 KB

    setup_minv_kernel<<<1, 64, 0, stream>>>(Minv);
    theta_prep_kernel<<<BATCH, 64, 0, stream>>>(traj_rot, out);
    theta_gemm_kernel<<<BATCH / 16, 32, 0, stream>>>(Minv, out);
    v_solve_kernel<<<BATCH / 32, 32, 0, stream>>>(traj_xy, v0, out);
}